// SimpleRNN_8967891714572
// MI455X (gfx1250) — compile-verified
//
#include <hip/hip_runtime.h>
#include <math.h>

#define T_STEPS 256
#define BATCH   128
#define INP_D   1024
#define HS_D    1024
#define OUT_D   1024

// LDS staging geometry: W chunk stored in WMMA B-fragment order.
// frag[g][kpair][n][2] : g = k-group of 4 (8 per 32-chunk), kpair = (k&3)>>1,
// n = 0..31 panel column, innermost = {B[kb][n], B[kb+1][n]} pair.
#define KC       32              // K-chunk depth
#define GRP_F    192             // floats per k-group: 2 kpairs * 96
#define KPAIR_F  96              // kpair stride: 96 % 64 == 32 banks -> conflict-free halves
#define CHUNK_F  (8 * GRP_F)     // 1536 floats per staged chunk

typedef __attribute__((ext_vector_type(2))) float v2f;
typedef __attribute__((ext_vector_type(8))) float v8f;

__device__ __forceinline__ v8f wmma_f32(v2f a, v2f b, v8f c) {
    // V_WMMA_F32_16X16X4_F32 : D = A(16x4) * B(4x16) + C(16x16)
    return __builtin_amdgcn_wmma_f32_16x16x4_f32(
        /*neg_a=*/false, a, /*neg_b=*/false, b,
        /*c_mod=*/(short)0, c, /*reuse_a=*/false, /*reuse_b=*/false);
}

// Workgroup barrier that only drains the DS counter (LDS stores must be
// visible) but leaves outstanding *global* loads in flight, unlike
// __syncthreads() which forces s_wait_loadcnt_dscnt 0. This keeps the
// double-buffer global prefetch overlapped with compute.
__device__ __forceinline__ void barrier_lds() {
    asm volatile(
        "s_wait_dscnt 0x0\n\t"
        "s_barrier_signal -1\n\t"
        "s_barrier_wait -1"
        ::: "memory");
}

__global__ void rnn_zero_kernel(float* __restrict__ p, int n) {
    int i = blockIdx.x * blockDim.x + threadIdx.x;
    if (i < n) p[i] = 0.0f;
}

// Scatter one loader thread's float4 (row k of chunk, cols lcol..lcol+3)
// into fragment order.
__device__ __forceinline__ void stage_frag(float* buf, int lrow, int lcol, float4 v) {
    const int g  = lrow >> 2;          // k-group
    const int hk = (lrow >> 1) & 1;    // kpair within group
    const int e  = lrow & 1;           // element within pair
    float* dst = buf + g * GRP_F + hk * KPAIR_F + e;
    dst[(lcol + 0) * 2] = v.x;
    dst[(lcol + 1) * 2] = v.y;
    dst[(lcol + 2) * 2] = v.z;
    dst[(lcol + 3) * 2] = v.w;
}

// Consume one staged 32-K chunk: load ALL fragments first (8 A from global,
// 16 B from LDS as 8x ds_load_2addr_b64-equivalent), then the 16-WMMA chain.
// Keeps ~24 loads in flight so waits become partial instead of full.
__device__ __forceinline__ void compute_chunk(
    const float* bufp, const float* ap, int half, int l16,
    v8f& acc0, v8f& acc1)
{
    v2f a[8], b0[8], b1[8];
    #pragma unroll
    for (int g = 0; g < 8; ++g) {
        a[g] = *(const v2f*)(ap + g * 4);        // {A[m][kb], A[m][kb+1]}
        const float* fb = bufp + g * GRP_F + half * KPAIR_F;
        b0[g] = *(const v2f*)(fb + l16 * 2);
        b1[g] = *(const v2f*)(fb + (l16 + 16) * 2);
    }
    #pragma unroll
    for (int g = 0; g < 8; ++g) {
        acc0 = wmma_f32(a[g], b0[g], acc0);
        acc1 = wmma_f32(a[g], b1[g], acc1);
    }
}

// Chunked GEMM accumulate: acc += A[m0:m0+16][:] * W[:][n0:n0+32]
// W panel staged through double-buffered LDS in fragment order; shared by
// all 8 waves (same n0, different m0). Last iteration peeled so the
// steady-state loop body is branch-free.
__device__ __forceinline__ void gemm_chunked(
    const float* __restrict__ A, int lda, int KTOT,
    const float* __restrict__ W,              // [KTOT][HS_D] row-major
    int m0, int n0, float* Bs,                // Bs: 2 * CHUNK_F floats
    v8f& acc0, v8f& acc1)
{
    const int tid  = threadIdx.x;
    const int lane = tid & 31;
    const int half = lane >> 4;               // 0: lanes 0-15, 1: lanes 16-31
    const int l16  = lane & 15;
    const int NC   = KTOT / KC;

    // cooperative loader mapping: 256 threads cover a KC x 32 chunk as float4
    const int lrow = tid >> 3;                // 0..31  (k within chunk)
    const int lcol = (tid & 7) << 2;          // 0,4,...,28 (n within panel)
    const float* gsrc   = W + (size_t)lrow * HS_D + n0 + lcol;
    const float* a_base = A + (size_t)(m0 + l16) * lda + half * 2;

    // prologue: fetch + stage chunk 0 into buffer 0
    float4 breg = *(const float4*)gsrc;
    stage_frag(Bs, lrow, lcol, breg);

    int c = 0;
    for (; c < NC - 1; ++c) {
        // prefetch chunk c+1 (stays in flight across the barrier)
        breg = *(const float4*)(gsrc + (size_t)(c + 1) * KC * HS_D);
        barrier_lds();                         // chunk c staged for all waves
        compute_chunk(Bs + (c & 1) * CHUNK_F, a_base + c * KC, half, l16,
                      acc0, acc1);
        // stage chunk c+1 into the other buffer (all threads passed the
        // barrier, so nobody still reads it from chunk c-1)
        stage_frag(Bs + ((c + 1) & 1) * CHUNK_F, lrow, lcol, breg);
    }
    barrier_lds();
    compute_chunk(Bs + (c & 1) * CHUNK_F, a_base + c * KC, half, l16,
                  acc0, acc1);
    barrier_lds();                             // Bs reusable by caller
}

// One recurrence step: h_out = tanh(x @ W_ih + h_in @ W_hh + b_i + b_h)
// Block = 256 threads = 8 waves; wave w owns batch rows [16w,16w+16),
// block owns hidden cols [32*blockIdx.x, +32). Grid = HS_D/32 = 32 blocks.
__global__ __launch_bounds__(256) void rnn_step_kernel(
    const float* __restrict__ x,      // [BATCH][INP_D] (slice for this t)
    const float* __restrict__ h_in,   // [BATCH][HS_D]
    const float* __restrict__ W_ih,   // [INP_D][HS_D]
    const float* __restrict__ W_hh,   // [HS_D][HS_D]
    const float* __restrict__ b_i,    // [HS_D]
    const float* __restrict__ b_h,    // [HS_D]
    float* __restrict__ h_out)        // [BATCH][HS_D]
{
    __shared__ float Bs[2 * CHUNK_F];

    const int wave = threadIdx.x >> 5;
    const int lane = threadIdx.x & 31;
    const int half = lane >> 4;
    const int l16  = lane & 15;

    const int m0 = wave * 16;
    const int n0 = blockIdx.x * 32;

    v8f acc0 = {};
    v8f acc1 = {};

    gemm_chunked(x,    INP_D, INP_D, W_ih, m0, n0, Bs, acc0, acc1);
    gemm_chunked(h_in, HS_D,  HS_D,  W_hh, m0, n0, Bs, acc0, acc1);

    const float bias0 = b_i[n0 + l16]      + b_h[n0 + l16];
    const float bias1 = b_i[n0 + 16 + l16] + b_h[n0 + 16 + l16];

    // C/D layout: VGPR r -> M = r + 8*half, N = l16
    #pragma unroll
    for (int r = 0; r < 8; ++r) {
        const int m = m0 + r + half * 8;
        h_out[(size_t)m * HS_D + n0 + l16]      = tanhf(acc0[r] + bias0);
        h_out[(size_t)m * HS_D + n0 + 16 + l16] = tanhf(acc1[r] + bias1);
    }
}

// Final projection: out = h @ W_out^T + b_out.
// B[k][n] = W_out[n][k] => per-lane contiguous float2 fragments from global.
__global__ __launch_bounds__(256) void rnn_out_kernel(
    const float* __restrict__ h,      // [BATCH][HS_D]
    const float* __restrict__ W_out,  // [OUT_D][HS_D]
    const float* __restrict__ b_out,  // [OUT_D]
    float* __restrict__ out)          // [BATCH][OUT_D]
{
    const int wave = threadIdx.x >> 5;
    const int lane = threadIdx.x & 31;
    const int half = lane >> 4;
    const int l16  = lane & 15;

    const int m0 = wave * 16;
    const int n0 = blockIdx.x * 32;

    v8f acc0 = {};
    v8f acc1 = {};

    const float* a_ptr  = h + (size_t)(m0 + l16) * HS_D + half * 2;
    const float* bbase0 = W_out + (size_t)(n0 + l16) * HS_D + half * 2;
    const float* bbase1 = W_out + (size_t)(n0 + 16 + l16) * HS_D + half * 2;

    #pragma unroll 4
    for (int k = 0; k < HS_D; k += 4) {
        v2f a  = *(const v2f*)(a_ptr + k);
        v2f b0 = *(const v2f*)(bbase0 + k);
        v2f b1 = *(const v2f*)(bbase1 + k);
        acc0 = wmma_f32(a, b0, acc0);
        acc1 = wmma_f32(a, b1, acc1);
    }

    const float bo0 = b_out[n0 + l16];
    const float bo1 = b_out[n0 + 16 + l16];

    #pragma unroll
    for (int r = 0; r < 8; ++r) {
        const int m = m0 + r + half * 8;
        out[(size_t)m * OUT_D + n0 + l16]      = acc0[r] + bo0;
        out[(size_t)m * OUT_D + n0 + 16 + l16] = acc1[r] + bo1;
    }
}

extern "C" void kernel_launch(void* const* d_in, const int* in_sizes, int n_in,
                              void* d_out, int out_size, void* d_ws, size_t ws_size,
                              hipStream_t stream) {
    (void)in_sizes; (void)n_in; (void)out_size; (void)ws_size;

    const float* data  = (const float*)d_in[0];  // [T][B][INP]
    const float* W_ih  = (const float*)d_in[1];  // [INP][HS]
    const float* W_hh  = (const float*)d_in[2];  // [HS][HS]
    const float* b_i   = (const float*)d_in[3];  // [HS]
    const float* b_h   = (const float*)d_in[4];  // [HS]
    const float* W_out = (const float*)d_in[5];  // [OUT][HS]
    const float* b_out = (const float*)d_in[6];  // [OUT]
    float* out = (float*)d_out;                  // [B][OUT]

    float* h0 = (float*)d_ws;                    // ping
    float* h1 = h0 + (size_t)BATCH * HS_D;       // pong

    const int hn = BATCH * HS_D;
    rnn_zero_kernel<<<(hn + 255) / 256, 256, 0, stream>>>(h0, hn);

    for (int t = 0; t < T_STEPS; ++t) {
        const float* hin  = (t & 1) ? h1 : h0;
        float*       hout = (t & 1) ? h0 : h1;
        rnn_step_kernel<<<HS_D / 32, 256, 0, stream>>>(
            data + (size_t)t * BATCH * INP_D, hin, W_ih, W_hh, b_i, b_h, hout);
    }
    // T_STEPS even -> final hidden state lives in h0
    rnn_out_kernel<<<HS_D / 32, 256, 0, stream>>>(h0, W_out, b_out, out);
}